// NeighborhoodSimilarityLayer_73478300500485
// MI455X (gfx1250) — compile-verified
//
#include <hip/hip_runtime.h>
#include <stdint.h>

#define B_ 4
#define C_ 64
#define H_ 256
#define W_ 256
#define TILE 16
#define HALO 1
#define TD (TILE + 2 * HALO)      // 18
#define LSTRIDE 24                // padded LDS row stride (floats) -> 96B rows
#define JOFF 3                    // logical jj 0..17 -> physical col JOFF..JOFF+17
#define CCHUNK 16
#define NCHUNK (C_ / CCHUNK)      // 4
#define EPSV 1e-8f

typedef float v2f __attribute__((ext_vector_type(2)));
typedef float v8f __attribute__((ext_vector_type(8)));

// ---------------- Kernel 1: per-(b,c) spatial mean via WMMA reduction ----------------
// D = ones(16x4) * B(4x16) + C sums 64 elements per v_wmma_f32_16x16x4_f32.
// With A == all-ones, sum(all entries of D) == 16 * sum(all elements fed through B),
// independent of the VGPR<->matrix element mapping, so the reduction is exact
// (the x16 replication divides out by a power of two).
__global__ void nsl_mean_kernel(const float* __restrict__ x, float* __restrict__ means) {
    __shared__ float red[256];
    const int bc   = blockIdx.x;        // 0..B*C-1
    const int tid  = threadIdx.x;
    const int wave = tid >> 5;          // 0..7
    const int lane = tid & 31;

    // 65536 floats per (b,c); 8192 per wave; 128 per wave-iteration (float4/lane)
    const float4* p = (const float4*)(x + (size_t)bc * (H_ * W_) + (size_t)wave * (H_ * W_ / 8));

    v2f ones; ones[0] = 1.0f; ones[1] = 1.0f;
    v8f acc0 = {}; v8f acc1 = {};

    for (int it = 0; it < (H_ * W_ / 8) / 128; ++it) {   // 64 iterations
        const float4 v = p[it * 32 + lane];
        v2f b0; b0[0] = v.x; b0[1] = v.y;
        v2f b1; b1[0] = v.z; b1[1] = v.w;
        acc0 = __builtin_amdgcn_wmma_f32_16x16x4_f32(false, ones, false, b0,
                                                     (short)0, acc0, false, false);
        acc1 = __builtin_amdgcn_wmma_f32_16x16x4_f32(false, ones, false, b1,
                                                     (short)0, acc1, false, false);
    }

    float s = 0.f;
#pragma unroll
    for (int k = 0; k < 8; ++k) s += acc0[k] + acc1[k];
    red[tid] = s;
    __syncthreads();
    for (int off = 128; off > 0; off >>= 1) {
        if (tid < off) red[tid] += red[tid + off];
        __syncthreads();
    }
    if (tid == 0) means[bc] = red[0] * (1.0f / (16.0f * (float)(H_ * W_)));
}

// ---------------- Kernel 2: neighborhood cosine similarity ----------------
// One 16x16 output tile per workgroup; channels staged through LDS in chunks of
// 16 using CDNA5 async global->LDS copies (ASYNCcnt). Interior 16 columns of
// every halo row are never reflected (tiles are 16-aligned in j), so they go
// through b128 async copies; only the two edge columns need reflected b32 copies.
__global__ void nsl_sim_kernel(const float* __restrict__ x,
                               const float* __restrict__ means,
                               float* __restrict__ out) {
    __shared__ float tile[CCHUNK][TD][LSTRIDE];  // 16*18*24*4 = 27648 B
    __shared__ float smean[C_];

    const int tj0 = blockIdx.x * TILE;
    const int ti0 = blockIdx.y * TILE;
    const int b   = blockIdx.z;
    const int tid = threadIdx.x;
    const int ti  = tid >> 4;    // 0..15
    const int tj  = tid & 15;    // 0..15

    if (tid < C_) smean[tid] = means[b * C_ + tid];

    const float* xb = x + (size_t)b * C_ * H_ * W_;

    float norm = 0.f;
    float dot[8], nn[8];
#pragma unroll
    for (int n = 0; n < 8; ++n) { dot[n] = 0.f; nn[n] = 0.f; }

    // neighbor order matches reference loop (di,dj) skipping center
    const int di_[8] = {-1, -1, -1, 0, 0, 1, 1, 1};
    const int dj_[8] = {-1, 0, 1, -1, 1, -1, 0, 1};

    const uint32_t lds_base = (uint32_t)(uintptr_t)(&tile[0][0][0]);

    for (int ch = 0; ch < NCHUNK; ++ch) {
        __syncthreads();  // previous chunk's readers done (also covers smean init)

        // ---- b128 interior: CCHUNK*TD rows x 4 vec4 = 1152 copies ----
        for (int e = tid; e < CCHUNK * TD * 4; e += 256) {
            const int cl = e >> 6;               // e / 72... (TD*4 = 72) -> use div
            const int cl2 = e / (TD * 4);
            const int r  = e - cl2 * (TD * 4);
            const int ii = r >> 2;
            const int q  = r & 3;
            (void)cl;
            int gi = ti0 - HALO + ii;
            gi = (gi < 0) ? -gi : ((gi >= H_) ? (2 * H_ - 2 - gi) : gi);
            const int c = ch * CCHUNK + cl2;
            const float* gp = xb + ((size_t)c * H_ + gi) * W_ + tj0 + q * 4;
            const uint32_t lds_off =
                lds_base + (uint32_t)(((cl2 * TD + ii) * LSTRIDE + (JOFF + 1) + q * 4) * 4);
            asm volatile("global_load_async_to_lds_b128 %0, %1, off"
                         :: "v"(lds_off), "v"(gp)
                         : "memory");
        }
        // ---- b32 edges: CCHUNK*TD rows x 2 = 576 copies (reflected j) ----
        for (int e = tid; e < CCHUNK * TD * 2; e += 256) {
            const int cl = e / (TD * 2);
            const int r  = e - cl * (TD * 2);
            const int ii = r >> 1;
            const int side = r & 1;
            int gi = ti0 - HALO + ii;
            gi = (gi < 0) ? -gi : ((gi >= H_) ? (2 * H_ - 2 - gi) : gi);
            int gj = side ? (tj0 + TILE) : (tj0 - 1);
            gj = (gj < 0) ? -gj : ((gj >= W_) ? (2 * W_ - 2 - gj) : gj);
            const int c = ch * CCHUNK + cl;
            const float* gp = xb + ((size_t)c * H_ + gi) * W_ + gj;
            const int jphys = side ? (JOFF + TD - 1) : JOFF;
            const uint32_t lds_off =
                lds_base + (uint32_t)(((cl * TD + ii) * LSTRIDE + jphys) * 4);
            asm volatile("global_load_async_to_lds_b32 %0, %1, off"
                         :: "v"(lds_off), "v"(gp)
                         : "memory");
        }
        // wait for this wave's async copies, then make them visible to all waves
        asm volatile("s_wait_asynccnt 0x0" ::: "memory");
        __syncthreads();

        // -------- accumulate over this chunk's channels --------
#pragma unroll
        for (int cl = 0; cl < CCHUNK; ++cl) {
            const float m  = smean[ch * CCHUNK + cl];
            const float cx = tile[cl][ti + 1][JOFF + tj + 1] - m;
            norm = fmaf(cx, cx, norm);
#pragma unroll
            for (int n = 0; n < 8; ++n) {
                const float cn = tile[cl][ti + 1 + di_[n]][JOFF + tj + 1 + dj_[n]] - m;
                dot[n] = fmaf(cn, cx, dot[n]);
                nn[n]  = fmaf(cn, cn, nn[n]);
            }
        }
    }

    const int gi = ti0 + ti;
    const int gj = tj0 + tj;
#pragma unroll
    for (int n = 0; n < 8; ++n) {
        const float den = __builtin_sqrtf(nn[n] * norm);
        const float sim = dot[n] / fmaxf(den, EPSV);
        out[(((size_t)b * 8 + n) * H_ + gi) * W_ + gj] = sim;
    }
}

extern "C" void kernel_launch(void* const* d_in, const int* in_sizes, int n_in,
                              void* d_out, int out_size, void* d_ws, size_t ws_size,
                              hipStream_t stream) {
    (void)in_sizes; (void)n_in; (void)out_size; (void)ws_size;
    const float* x   = (const float*)d_in[0];
    float* out       = (float*)d_out;
    float* means     = (float*)d_ws;   // 256 floats of scratch

    hipLaunchKernelGGL(nsl_mean_kernel, dim3(B_ * C_), dim3(256), 0, stream, x, means);
    hipLaunchKernelGGL(nsl_sim_kernel, dim3(W_ / TILE, H_ / TILE, B_), dim3(256), 0, stream,
                       x, means, out);
}